// DfOpTimeLoop_30185030156321
// MI455X (gfx1250) — compile-verified
//
#include <hip/hip_runtime.h>

// DfOp time loop for MI455X (gfx1250).
// Memory-bound (~0.7 GB traffic, ~0.27 GFLOP) -> optimize data movement.
//  * CDNA5 async global->LDS loads (ASYNCcnt path) stage the 5-tap spec window
//    once per block; tail copy overlaps the async latency before the wait.
//  * Non-temporal hints on all zero-reuse streams (coefs, tail, output).

#define NF   481   // NFREQ
#define NDF  96
#define ORD  5
#define LA   2     // LOOKAHEAD
#define TB   32    // time frames per block
#define NROWS (TB + 2 * LA)   // 36 staged rows max
#define THREADS 256

typedef float f2 __attribute__((ext_vector_type(2)));

__global__ __launch_bounds__(THREADS)
void dfop_kernel(const f2* __restrict__ spec,
                 const f2* __restrict__ coefs,
                 const float* __restrict__ alpha,
                 f2* __restrict__ out,
                 int T)
{
    __shared__ f2 srow[NROWS][NDF];   // 27,648 bytes

    const int t0  = blockIdx.x * TB;
    const int tid = threadIdx.x;
    const int r0   = (t0 - LA > 0) ? (t0 - LA) : 0;             // first staged row
    const int rend = ((t0 + TB + LA) < T) ? (t0 + TB + LA) : T; // exclusive
    const int nr   = rend - r0;                                 // <= NROWS

    // ---- Issue async stage of spec[r0:rend, 0:NDF] into LDS (b64, 8B-aligned rows) ----
    {
        const unsigned lbase = (unsigned)(size_t)(&srow[0][0]); // LDS byte offset
        const int total = nr * NDF;
        for (int i = tid; i < total; i += THREADS) {
            const int r = i / NDF;
            const int c = i - r * NDF;
            const unsigned lds_off = lbase + (unsigned)(i * sizeof(f2));
            const unsigned long long gaddr =
                (unsigned long long)(size_t)(spec + (size_t)(r0 + r) * NF + c);
            asm volatile("global_load_async_to_lds_b64 %0, %1, off"
                         :: "v"(lds_off), "v"(gaddr) : "memory");
        }
    }

    // ---- Pass-through tail (f in [NDF, NF)) while async loads are in flight ----
    const int NTAIL = NF - NDF;   // 385
    for (int i = tid; i < TB * NTAIL; i += THREADS) {
        const int lt = i / NTAIL;
        const int f  = NDF + (i - lt * NTAIL);
        const int gt = t0 + lt;
        if (gt >= T) break;
        const int bidx = (gt >= LA) ? gt : (1 - gt);
        const f2 v = __builtin_nontemporal_load(&spec[(size_t)bidx * NF + f]);
        __builtin_nontemporal_store(v, &out[(size_t)gt * NF + f]);
    }

    // ---- Complete staging, make LDS visible to all waves ----
    asm volatile("s_wait_asynccnt 0" ::: "memory");
    __syncthreads();

    // ---- Filtered + blended region: f in [0, NDF) ----
    for (int i = tid; i < TB * NDF; i += THREADS) {
        const int lt = i / NDF;           // wave-uniform (NDF % 32 == 0)
        const int f  = i - lt * NDF;
        const int gt = t0 + lt;
        if (gt >= T) break;

        float accr = 0.0f, acci = 0.0f;
        #pragma unroll
        for (int j = 0; j < ORD; ++j) {
            const int k = ORD - 1 - j;
            int idx, m;
            if (k <= gt) { idx = gt + LA - k; m = (idx < T); } // zero-pad rows masked
            else         { idx = k - gt - 1;  m = (idx < LA); }
            if (m) {
                const f2 s = srow[idx - r0][f];
                const f2 c = __builtin_nontemporal_load(
                                 &coefs[((size_t)gt * ORD + j) * NDF + f]);
                accr += s.x * c.x - s.y * c.y;
                acci += s.x * c.y + s.y * c.x;
            }
        }
        const int bidx = (gt >= LA) ? gt : (1 - gt);   // base mask always 1, in-range
        const f2 b = srow[bidx - r0][f];
        const float a = alpha[gt];
        f2 o;
        o.x = accr * a + b.x * (1.0f - a);
        o.y = acci * a + b.y * (1.0f - a);
        __builtin_nontemporal_store(o, &out[(size_t)gt * NF + f]);
    }
}

extern "C" void kernel_launch(void* const* d_in, const int* in_sizes, int n_in,
                              void* d_out, int out_size, void* d_ws, size_t ws_size,
                              hipStream_t stream) {
    (void)n_in; (void)out_size; (void)d_ws; (void)ws_size;
    const f2* spec     = (const f2*)d_in[0];    // (T, 481, 2) f32
    const f2* coefs    = (const f2*)d_in[1];    // (T, 5, 96, 2) f32
    const float* alpha = (const float*)d_in[2]; // (T, 1) f32
    f2* out = (f2*)d_out;                       // (T, 481, 2) f32

    const int T = in_sizes[0] / (NF * 2);
    const int grid = (T + TB - 1) / TB;
    hipLaunchKernelGGL(dfop_kernel, dim3(grid), dim3(THREADS), 0, stream,
                       spec, coefs, alpha, out, T);
}